// HyperbolicMarginLoss_49203145343004
// MI455X (gfx1250) — compile-verified
//
#include <hip/hip_runtime.h>
#include <hip/hip_bf16.h>
#include <math.h>

// HyperbolicMarginLoss for MI455X (gfx1250), wave32, fp32 WMMA path.
// B=8192, D=128, N_CLASSES=256, CURV=1, MARGIN=0.5

typedef float v2f __attribute__((ext_vector_type(2)));
typedef float v8f __attribute__((ext_vector_type(8)));

#define HML_B 8192
#define HML_D 128
#define HML_NT (HML_B / 16)      // 512 column tiles
#define HML_SPLIT 8              // column chunks -> 8x more blocks for fill
#define HML_TPC (HML_NT / HML_SPLIT)  // 64 tiles per chunk
#define LDSTR 132                // 128 + 4 pad floats -> conflict-free ds_load_b64

#define POS_SENTINEL (-1.0f)
#define NEG_SENTINEL (3.0e38f)

// ---------------------------------------------------------------------------
// Kernel 1: per-row squared norm, inv = 1/(1-sq); init per-row max/min arrays.
// One 32-lane wave per row; 8 rows per 256-thread block.
// ---------------------------------------------------------------------------
__global__ __launch_bounds__(256) void hml_sqnorm(const float* __restrict__ X,
                                                  float* __restrict__ sq,
                                                  float* __restrict__ inv,
                                                  float* __restrict__ pmaxA,
                                                  float* __restrict__ pminA) {
  const int row  = blockIdx.x * 8 + (threadIdx.x >> 5);
  const int lane = threadIdx.x & 31;
  const float4 x = *(const float4*)(X + (size_t)row * HML_D + lane * 4);
  float s = x.x * x.x + x.y * x.y + x.z * x.z + x.w * x.w;
#pragma unroll
  for (int m = 16; m >= 1; m >>= 1) s += __shfl_xor(s, m, 32);
  if (lane == 0) {
    sq[row]    = s;
    inv[row]   = 1.0f / (1.0f - s);   // norms < 0.8 -> always well defined
    pmaxA[row] = POS_SENTINEL;
    pminA[row] = NEG_SENTINEL;
  }
}

// ---------------------------------------------------------------------------
// Kernel 2: tiled X*X^T via V_WMMA_F32_16X16X4_F32 with fused masked
// row-max/row-min reduction of the (monotone) hyperbolic distance argument.
// Block = 128 threads (4 waves); each wave owns a 16-row stripe with its A
// fragments resident in 64 VGPRs; block loops over one 64-tile column chunk
// staged through LDS; partial results merged with float atomic max/min.
// ---------------------------------------------------------------------------
__global__ __launch_bounds__(128) void hml_main(const float* __restrict__ X,
                                                const int*   __restrict__ labels,
                                                const float* __restrict__ sq,
                                                const float* __restrict__ inv,
                                                float* __restrict__ pmaxA,
                                                float* __restrict__ pminA) {
  __shared__ alignas(16) float bt[16 * LDSTR];

  const int tid  = threadIdx.x;
  const int lane = tid & 31;
  const int wave = tid >> 5;
  const int half = lane >> 4;     // 0: K%4 in {0,1}, 1: K%4 in {2,3}
  const int l15  = lane & 15;
  const int rowBase = blockIdx.x * 64 + wave * 16;

  // --- A fragments: 16 rows x 128 cols, kept in registers (32 x v2f) ------
  // A layout (16x4 f32): lane<16 holds {A[l15][k0], A[l15][k0+1]},
  //                      lane>=16 holds {A[l15][k0+2], A[l15][k0+3]}.
  const float* Arow = X + (size_t)(rowBase + l15) * HML_D + 2 * half;
  v2f a[32];
#pragma unroll
  for (int ks = 0; ks < 32; ++ks) a[ks] = *(const v2f*)(Arow + 4 * ks);

  // --- Row-side labels / sq / inv (C layout: VGPR v -> row v + 8*half) ----
  int   lr[8];
  float sqr[8], invr[8];
#pragma unroll
  for (int v = 0; v < 8; ++v) {
    const int r = rowBase + v + 8 * half;
    lr[v] = labels[r];  sqr[v] = sq[r];  invr[v] = inv[r];
  }

  float pmax[8], pmin[8];
#pragma unroll
  for (int v = 0; v < 8; ++v) { pmax[v] = POS_SENTINEL; pmin[v] = NEG_SENTINEL; }

  const int jt0 = blockIdx.y * HML_TPC;
  for (int jt = jt0; jt < jt0 + HML_TPC; ++jt) {
    const int J = jt * 16;

    // Cooperative stage of the 16x128 B tile into padded LDS (512 float4s).
#pragma unroll
    for (int i = 0; i < 4; ++i) {
      const int f  = tid + i * 128;      // float4 index in [0,512)
      const int r  = f >> 5;             // row 0..15
      const int kc = (f & 31) << 2;      // col 0,4,...,124
      *(float4*)(&bt[r * LDSTR + kc]) =
          *(const float4*)(X + (size_t)(J + r) * HML_D + kc);
    }
    __syncthreads();

    const int   lc   = labels[J + l15];
    const float sqc  = sq[J + l15];
    const float invc = inv[J + l15];

    v8f c = {};
    // B layout mirrors A (column per lane, K striped identically across
    // VGPR halves), so loading X[J+l15] rows like A gives B = X_J^T.
    const float* brow = bt + l15 * LDSTR + 2 * half;
#pragma unroll
    for (int ks = 0; ks < 32; ++ks) {
      v2f b = *(const v2f*)(brow + 4 * ks);
      c = __builtin_amdgcn_wmma_f32_16x16x4_f32(
              false, a[ks], false, b, (short)0, c, false, false);
    }
    __syncthreads();   // tile consumed; safe to overwrite next iteration

    // Epilogue: u = max(sq_i + sq_j - 2*g, 0) * invc   (invr folded out:
    // it is a positive row constant, applied once after the reduction).
    if (J != rowBase) {            // 511/512 tiles: no diagonal possible
#pragma unroll
      for (int v = 0; v < 8; ++v) {
        const float d2 = fmaxf(sqr[v] + sqc - 2.0f * c[v], 0.0f);
        const float u  = d2 * invc;
        if (lr[v] == lc) pmax[v] = fmaxf(pmax[v], u);
        else             pmin[v] = fminf(pmin[v], u);
      }
    } else {                       // tile containing the diagonal
#pragma unroll
      for (int v = 0; v < 8; ++v) {
        const float d2 = fmaxf(sqr[v] + sqc - 2.0f * c[v], 0.0f);
        const float u  = d2 * invc;
        const bool diag = ((v + 8 * half) == l15);
        if (lr[v] == lc) { if (!diag) pmax[v] = fmaxf(pmax[v], u); }
        else             pmin[v] = fminf(pmin[v], u);
      }
    }
  }

  // Reduce across the 16 lanes of each half (rows v+0 in half0, v+8 in half1).
#pragma unroll
  for (int v = 0; v < 8; ++v) {
#pragma unroll
    for (int m = 8; m >= 1; m >>= 1) {
      pmax[v] = fmaxf(pmax[v], __shfl_xor(pmax[v], m, 32));
      pmin[v] = fminf(pmin[v], __shfl_xor(pmin[v], m, 32));
    }
  }

  if (l15 == 0) {  // lanes 0 and 16 each merge 8 rows into the global arrays
#pragma unroll
    for (int v = 0; v < 8; ++v) {
      const int r = rowBase + v + 8 * half;
      atomicMax(&pmaxA[r], pmax[v] * invr[v]);   // sentinel -1 stays negative
      atomicMin(&pminA[r], pmin[v] * invr[v]);   // sentinel stays >= 3e38
    }
  }
}

// ---------------------------------------------------------------------------
// Kernel 3: per-row acosh + margin, reduce to scalar. Single 256-thread block.
// ---------------------------------------------------------------------------
__global__ __launch_bounds__(256) void hml_finalize(const float* __restrict__ pmaxA,
                                                    const float* __restrict__ pminA,
                                                    float* __restrict__ out) {
  __shared__ float ssum[8], scnt[8];
  const int tid  = threadIdx.x;
  const int lane = tid & 31;
  const int wave = tid >> 5;

  float lsum = 0.0f, lcnt = 0.0f;
  for (int r = tid; r < HML_B; r += 256) {
    const float pm = pmaxA[r];
    const float pn = pminA[r];
    const bool hp = (pm >= 0.0f);
    const bool hn = (pn < 1.0e37f);
    if (hp && hn) {
      const float dpos = acoshf(fmaxf(1.0f + 2.0f * pm, 1.0f + 1e-7f));
      const float dneg = acoshf(fmaxf(1.0f + 2.0f * pn, 1.0f + 1e-7f));
      lsum += fmaxf(dpos - dneg + 0.5f, 0.0f);
      lcnt += 1.0f;
    }
  }
#pragma unroll
  for (int m = 16; m >= 1; m >>= 1) {
    lsum += __shfl_xor(lsum, m, 32);
    lcnt += __shfl_xor(lcnt, m, 32);
  }
  if (lane == 0) { ssum[wave] = lsum; scnt[wave] = lcnt; }
  __syncthreads();
  if (tid == 0) {
    float s = 0.0f, c = 0.0f;
#pragma unroll
    for (int w = 0; w < 8; ++w) { s += ssum[w]; c += scnt[w]; }
    out[0] = (c > 0.0f) ? (s / c) : 0.0f;
  }
}

extern "C" void kernel_launch(void* const* d_in, const int* in_sizes, int n_in,
                              void* d_out, int out_size, void* d_ws, size_t ws_size,
                              hipStream_t stream) {
  (void)in_sizes; (void)n_in; (void)out_size; (void)ws_size;
  const float* X      = (const float*)d_in[0];
  const int*   labels = (const int*)d_in[1];   // jnp.int64 -> int32 under default JAX x64-off
  float* out = (float*)d_out;

  float* w     = (float*)d_ws;
  float* pmaxA = w;                  // 8192 floats
  float* pminA = w + HML_B;          // 8192 floats
  float* sq    = w + 2 * HML_B;      // 8192 floats
  float* inv   = w + 3 * HML_B;      // 8192 floats

  hml_sqnorm<<<HML_B / 8, 256, 0, stream>>>(X, sq, inv, pmaxA, pminA);
  dim3 grid(HML_B / 64, HML_SPLIT);
  hml_main<<<grid, 128, 0, stream>>>(X, labels, sq, inv, pmaxA, pminA);
  hml_finalize<<<1, 256, 0, stream>>>(pmaxA, pminA, out);
}